// YoloLoss_335007450062
// MI455X (gfx1250) — compile-verified
//
#include <hip/hip_runtime.h>
#include <math.h>

#define NB 32
#define NP 100
#define NA 25200
#define NC 80
#define EPSF 1e-7f
#define POS_THRF 0.5f
#define NEG_THRF 0.4f
#define ALPHAF 0.25f

typedef __attribute__((ext_vector_type(2))) float v2f;
typedef __attribute__((ext_vector_type(8))) float v8f;

// ---------------------------------------------------------------------------
// Wave32 sum-reduction using V_WMMA_F32_16X16X4_F32.
// A-matrix layout (16x4 f32): VGPR0 lane m (m<16) = A[m][0], lane 16+m = A[m][2].
// We place x[lane] into A VGPR0 and zero VGPR1, B = all-ones:
//   D[m][n] = x[m] + x[m+16]  (= s_m, replicated over n)
// D VGPR j holds s_j on lanes 0-15 and s_{8+j} on lanes 16-31, so summing the
// 8 D VGPRs yields (x[0..7]+x[16..23]) on lanes 0-15 and (x[8..15]+x[24..31])
// on lanes 16-31: lane0 + lane16 = full wave sum.
// ---------------------------------------------------------------------------
__device__ __forceinline__ float wave_halfsum_wmma(float x) {
  v2f a; a.x = x;   a.y = 0.0f;
  v2f b; b.x = 1.0f; b.y = 1.0f;
  v8f c = {};
  c = __builtin_amdgcn_wmma_f32_16x16x4_f32(false, a, false, b,
                                            (short)0, c, false, false);
  return c[0] + c[1] + c[2] + c[3] + c[4] + c[5] + c[6] + c[7];
}

__device__ __forceinline__ float ciou_loss_dev(const float* bt, const float* bp) {
  float bt0 = bt[0], bt1 = bt[1], bt2 = bt[2], bt3 = bt[3];
  float bp0 = bp[0], bp1 = bp[1], bp2 = bp[2], bp3 = bp[3];
  float iw = fminf(bt2, bp2) - fmaxf(bt0, bp0);
  float ih = fminf(bt3, bp3) - fmaxf(bt1, bp1);
  float inter = fmaxf(iw, 0.0f) * fmaxf(ih, 0.0f);
  float wt = bt2 - bt0, ht = bt3 - bt1;
  float wp = bp2 - bp0, hp = bp3 - bp1;
  float uni = wt * ht + wp * hp - inter + EPSF;
  float iou = inter / uni;
  float cw = fmaxf(bt2, bp2) - fminf(bt0, bp0);
  float ch = fmaxf(bt3, bp3) - fminf(bt1, bp1);
  float c2 = cw * cw + ch * ch + EPSF;
  float dx = bt0 + bt2 - bp0 - bp2;
  float dy = bt1 + bt3 - bp1 - bp3;
  float rho2 = (dx * dx + dy * dy) * 0.25f;
  float dv = atanf(wt / (ht + EPSF)) - atanf(wp / (hp + EPSF));
  const float kk = 4.0f / (3.14159265358979323846f * 3.14159265358979323846f);
  float v = kk * dv * dv;
  float alpha = v / (1.0f - iou + v + EPSF);
  return 1.0f - iou + rho2 / c2 + alpha * v;
}

// ---------------------------------------------------------------------------
// Pass 0: zero the per-GT argmax table and the accumulators.
// ---------------------------------------------------------------------------
__global__ void init_kernel(unsigned long long* __restrict__ packed,
                            float* __restrict__ accum) {
  int i = blockIdx.x * blockDim.x + threadIdx.x;
  if (i < NB * NP) packed[i] = 0ull;
  if (i < 64) accum[i] = 0.0f;
}

// ---------------------------------------------------------------------------
// Pass 1: anchor assignment. One thread per (n, anchor).
//   assign[n*A+a] = gt_idx (bits 7:0) | state<<8  (state: 0=neg,1=neutral,2=pos)
//   packed[n*P+p] = atomicMax of (iou_bits<<32)|(~anchor) over anchors (valid p)
// ---------------------------------------------------------------------------
__global__ void assign_kernel(const float* __restrict__ bbox_true,
                              const float* __restrict__ anchors,
                              unsigned* __restrict__ assign,
                              unsigned long long* __restrict__ packed) {
  __shared__ float4 gtBox[NP];
  __shared__ float  gtArea[NP];
  __shared__ int    gtValid[NP];

  const int n   = blockIdx.y;
  const int tid = threadIdx.x;
  if (tid < NP) {
    const float* g = bbox_true + ((size_t)n * NP + tid) * 4;
    float4 b = make_float4(g[0], g[1], g[2], g[3]);
    gtBox[tid]  = b;
    gtArea[tid] = (b.z - b.x) * (b.w - b.y);
    gtValid[tid] = (b.x > 0.0f) || (b.y > 0.0f) || (b.z > 0.0f) || (b.w > 0.0f);
  }
  __syncthreads();

  const int a = blockIdx.x * blockDim.x + tid;
  const bool live = (a < NA);
  float ax1 = 0.f, ay1 = 0.f, ax2 = 0.f, ay2 = 0.f, areaA = 0.f;
  if (live) {
    const float* ap = anchors + (size_t)a * 4;
    ax1 = ap[0]; ay1 = ap[1]; ax2 = ap[2]; ay2 = ap[3];
    areaA = (ax2 - ax1) * (ay2 - ay1);
  }

  float best = -1.0f;  // matches where(valid, iou, -1)
  int   bidx = 0;
  for (int p = 0; p < NP; ++p) {
    if (!gtValid[p]) continue;   // uniform branch: invalid column never changes best
    float4 b = gtBox[p];
    float iw = fmaxf(fminf(ax2, b.z) - fmaxf(ax1, b.x), 0.0f);
    float ih = fmaxf(fminf(ay2, b.w) - fmaxf(ay1, b.y), 0.0f);
    float inter = iw * ih;
    float iou = inter / (areaA + gtArea[p] - inter + EPSF);
    if (live && iou > best) { best = iou; bidx = p; }  // strict > = first-index tie-break

    // GT-side argmax over anchors (packed key; smaller anchor idx wins ties)
    unsigned long long key = 0ull;
    if (live && iou > 0.0f) {
      key = ((unsigned long long)__float_as_uint(iou) << 32) |
            (unsigned long long)(0xFFFFFFFFu - (unsigned)a);
    }
    for (int off = 16; off > 0; off >>= 1) {
      unsigned long long o = __shfl_xor(key, off, 32);
      if (o > key) key = o;
    }
    if ((tid & 31) == 0 && key != 0ull) {
      atomicMax(&packed[n * NP + p], key);
    }
  }

  if (live) {
    int state = (best >= POS_THRF) ? 2 : ((best < NEG_THRF) ? 0 : 1);
    assign[(size_t)n * NA + a] = (unsigned)bidx | ((unsigned)state << 8);
  }
}

// ---------------------------------------------------------------------------
// Pass 2: low-quality match override. One thread per (n, gt).
// ---------------------------------------------------------------------------
__global__ void lowq_kernel(unsigned* __restrict__ assign,
                            const unsigned long long* __restrict__ packed) {
  int i = blockIdx.x * blockDim.x + threadIdx.x;
  if (i >= NB * NP) return;
  int n = i / NP;
  int p = i % NP;
  unsigned long long key = packed[i];
  unsigned hi = (unsigned)(key >> 32);
  if (hi == 0u) return;  // gt_max <= 0 (or invalid gt: never updated)
  unsigned a = 0xFFFFFFFFu - (unsigned)(key & 0xFFFFFFFFull);
  assign[(size_t)n * NA + a] = (unsigned)p | (2u << 8);
}

// ---------------------------------------------------------------------------
// Pass 3: losses. One thread per (n, anchor).  Focal loss over C=80 is done
// wave-cooperatively for each positive lane (positives are sparse).  Block
// reduction goes through v_wmma_f32_16x16x4_f32.
// accum layout: [0]=score_num [1]=cls_num [2]=bbox_num [8+n]=pos count for n
// ---------------------------------------------------------------------------
__global__ void loss_kernel(const float* __restrict__ y_true,
                            const float* __restrict__ bbox_true,
                            const float* __restrict__ conf_pred,
                            const float* __restrict__ logit_pred,
                            const float* __restrict__ bbox_pred,
                            const unsigned* __restrict__ assign,
                            float* __restrict__ accum) {
  __shared__ float lacc[4];
  const int tid = threadIdx.x;
  if (tid < 4) lacc[tid] = 0.0f;
  __syncthreads();

  const int n = blockIdx.y;
  const int a = blockIdx.x * blockDim.x + tid;
  const bool live = (a < NA);
  const long long na = (long long)n * NA + a;

  float score_acc = 0.0f, cls_acc = 0.0f, bbox_acc = 0.0f, cnt_acc = 0.0f;
  int idx = 0;
  bool pos = false;

  if (live) {
    unsigned w = assign[na];
    idx = (int)(w & 0xFFu);
    int st = (int)((w >> 8) & 3u);
    pos = (st == 2);
    bool neg = (st == 0);

    float pc = conf_pred[na];
    pc = fminf(fmaxf(pc, EPSF), 1.0f - EPSF);
    if (pos)      { score_acc = -logf(pc);        cnt_acc = 1.0f; }
    else if (neg) { score_acc = -logf(1.0f - pc); }

    if (pos) {
      const float* bt = bbox_true + ((size_t)n * NP + idx) * 4;
      const float* bp = bbox_pred + (size_t)na * 4;
      bbox_acc = ciou_loss_dev(bt, bp);
    }
  }

  // Wave-cooperative focal loss over the positive lanes.
  const int lane = tid & 31;
  unsigned mask = (unsigned)__ballot(pos);
  while (mask) {
    int src = __ffs(mask) - 1;
    mask &= mask - 1;
    int idx_s = __shfl(idx, src, 32);
    int a_s   = __shfl(a,   src, 32);
    const float* lrow = logit_pred + ((size_t)n * NA + a_s) * (size_t)NC;
    const float* trow = y_true    + ((size_t)n * NP + idx_s) * (size_t)NC;
    __builtin_prefetch(lrow + lane, 0, 0);   // global_prefetch_b8
    for (int c = lane; c < NC; c += 32) {
      float t = trow[c];
      float q = fminf(fmaxf(lrow[c], EPSF), 1.0f - EPSF);
      float pt = t * q + (1.0f - t) * (1.0f - q);
      float at = t * ALPHAF + (1.0f - t) * (1.0f - ALPHAF);
      float om = 1.0f - pt;
      cls_acc += -at * om * om * logf(pt);
    }
  }

  // Fully converged here: EXEC all-1s, legal for WMMA.
  float ts = wave_halfsum_wmma(score_acc);
  float tc = wave_halfsum_wmma(cls_acc);
  float tb = wave_halfsum_wmma(bbox_acc);
  float tn = wave_halfsum_wmma(cnt_acc);
  if (lane == 0 || lane == 16) {
    atomicAdd(&lacc[0], ts);
    atomicAdd(&lacc[1], tc);
    atomicAdd(&lacc[2], tb);
    atomicAdd(&lacc[3], tn);
  }
  __syncthreads();
  if (tid == 0) {
    atomicAdd(&accum[0], lacc[0]);
    atomicAdd(&accum[1], lacc[1]);
    atomicAdd(&accum[2], lacc[2]);
    atomicAdd(&accum[8 + n], lacc[3]);
  }
}

// ---------------------------------------------------------------------------
// Pass 4: avg_factor, division, NaN/Inf guard.
// ---------------------------------------------------------------------------
__global__ void finalize_kernel(const float* __restrict__ accum,
                                float* __restrict__ out) {
  if (blockIdx.x == 0 && threadIdx.x == 0) {
    float af = 0.0f;
    for (int nn = 0; nn < NB; ++nn) af += fmaxf(accum[8 + nn], 1.0f);
    for (int k = 0; k < 3; ++k) {
      float v = accum[k] / af;
      out[k] = (isnan(v) || isinf(v)) ? 0.0f : v;
    }
  }
}

extern "C" void kernel_launch(void* const* d_in, const int* in_sizes, int n_in,
                              void* d_out, int out_size, void* d_ws, size_t ws_size,
                              hipStream_t stream) {
  const float* y_true     = (const float*)d_in[0];  // (N,P,C)
  const float* bbox_true  = (const float*)d_in[1];  // (N,P,4)
  const float* conf_pred  = (const float*)d_in[2];  // (N,A,1)
  const float* logit_pred = (const float*)d_in[3];  // (N,A,C)
  const float* bbox_pred  = (const float*)d_in[4];  // (N,A,4)
  const float* anchors    = (const float*)d_in[5];  // (A,4)
  float* out = (float*)d_out;

  char* ws = (char*)d_ws;
  float*              accum  = (float*)ws;                          // 64 floats
  unsigned long long* packed = (unsigned long long*)(ws + 256);     // N*P u64
  unsigned*           assign = (unsigned*)(ws + 32768);             // N*A u32

  init_kernel<<<dim3((NB * NP + 255) / 256), 256, 0, stream>>>(packed, accum);

  dim3 grid((NA + 255) / 256, NB);
  assign_kernel<<<grid, 256, 0, stream>>>(bbox_true, anchors, assign, packed);

  lowq_kernel<<<dim3((NB * NP + 255) / 256), 256, 0, stream>>>(assign, packed);

  loss_kernel<<<grid, 256, 0, stream>>>(y_true, bbox_true, conf_pred,
                                        logit_pred, bbox_pred, assign, accum);

  finalize_kernel<<<1, 32, 0, stream>>>(accum, out);
}